// GraphModel_43336220016774
// MI455X (gfx1250) — compile-verified
//
#include <hip/hip_runtime.h>
#include <hip/hip_bf16.h>

typedef __attribute__((ext_vector_type(2))) float v2f;
typedef __attribute__((ext_vector_type(8))) float v8f;

#define D_CH 100
#define NHEAD 2
#define NEG_SLOPE 0.2f
#define KSTEPS (D_CH / 4)   // 25 WMMA K-steps of 4

// ---------------------------------------------------------------------------
// Monotonic float <-> uint32 key (total order) for atomicMax-based segment max
// ---------------------------------------------------------------------------
__device__ __forceinline__ unsigned fkey(float f) {
    unsigned b = __float_as_uint(f);
    return (b & 0x80000000u) ? ~b : (b | 0x80000000u);
}
__device__ __forceinline__ float funkey(unsigned k) {
    unsigned b = (k & 0x80000000u) ? (k & 0x7FFFFFFFu) : ~k;
    return __uint_as_float(b);
}
#define NEG_INF_KEY 0x007FFFFFu   // fkey(-inf)

// ---------------------------------------------------------------------------
// Kernel 1: fold W against att_src / att_dst / W_lin.
//   U[k, 0..1] = sum_d W[k, h*D+d] * att_src[h,d]   (h = 0,1)
//   U[k, 2..3] = sum_d W[k, h*D+d] * att_dst[h,d]
//   U[k, 4..5] = sum_d W[k, h*D+d] * W_lin[d]
//   U padded to 16 columns with zeros (WMMA tile width).
// Also computes c0 = bias . W_lin + b_lin (scalar constant of the output).
// ---------------------------------------------------------------------------
__global__ void prep_U_kernel(const float* __restrict__ W,
                              const float* __restrict__ att_src,
                              const float* __restrict__ att_dst,
                              const float* __restrict__ W_lin,
                              const float* __restrict__ bias,
                              const float* __restrict__ b_lin,
                              float* __restrict__ U,
                              float* __restrict__ c0) {
    int t = blockIdx.x * blockDim.x + threadIdx.x;
    if (t >= D_CH * 16) return;
    int k = t >> 4;
    int c = t & 15;
    float acc = 0.0f;
    if (c < 6) {
        int h = c & 1;
        const float* wrow = W + (size_t)k * (NHEAD * D_CH) + h * D_CH;
        if (c < 2) {
            const float* v = att_src + h * D_CH;
            for (int d = 0; d < D_CH; ++d) acc += wrow[d] * v[d];
        } else if (c < 4) {
            const float* v = att_dst + h * D_CH;
            for (int d = 0; d < D_CH; ++d) acc += wrow[d] * v[d];
        } else {
            for (int d = 0; d < D_CH; ++d) acc += wrow[d] * W_lin[d];
        }
    }
    U[k * 16 + c] = acc;
    if (t == 0) {
        float s = 0.0f;
        for (int d = 0; d < D_CH; ++d) s += bias[d] * W_lin[d];
        *c0 = s + b_lin[0];
    }
}

// ---------------------------------------------------------------------------
// Kernel 2: S = x @ U  via V_WMMA_F32_16X16X4_F32.
// One wave per PAIR of 16-row tiles (32 rows):
//   - B (the U matrix) is loop-invariant -> hoisted into 50 VGPRs once,
//     removing all B traffic + address math from the hot loop.
//   - Two independent accumulator chains interleave WMMAs so consecutive
//     v_wmma issues have no D->C RAW dependence.
// A (16x4 f32) lane layout: lanes 0-15 -> row M=lane, K = k0+0 (v0), k0+1 (v1)
//                           lanes 16-31 -> row M=lane-16, K = k0+2, k0+3
// C/D 16x16 f32: VGPR r = row M=r (lanes 0-15) / M=r+8 (lanes 16-31), col = lane%16.
// ---------------------------------------------------------------------------
__global__ __launch_bounds__(256) void node_gemm_wmma(const float* __restrict__ x,
                                                      const float* __restrict__ U,
                                                      float* __restrict__ S,
                                                      int nNodes) {
    const int lane = threadIdx.x & 31;
    const int wave = threadIdx.x >> 5;
    const int pair = blockIdx.x * 8 + wave;       // pair of 16-row tiles (wave-uniform)
    const int row0 = pair * 32;
    if (row0 >= nNodes) return;
    const int hf   = lane >> 4;                   // 0: K pair {0,1}, 1: K pair {2,3}
    const int l16  = lane & 15;

    // ---- hoist B into registers: 25 x v2f = 50 VGPRs, loop-invariant ----
    v2f bv[KSTEPS];
    #pragma unroll
    for (int i = 0; i < KSTEPS; ++i) {
        const int ka = i * 4 + 2 * hf;
        bv[i].x = U[(ka + 0) * 16 + l16];
        bv[i].y = U[(ka + 1) * 16 + l16];
    }

    int r0 = row0 + l16;        if (r0 > nNodes - 1) r0 = nNodes - 1;   // clamp:
    int r1 = row0 + 16 + l16;   if (r1 > nNodes - 1) r1 = nNodes - 1;   // EXEC all-1s
    const float* __restrict__ xa = x + (size_t)r0 * D_CH;
    const float* __restrict__ xb = x + (size_t)r1 * D_CH;

    v8f acc0 = {};
    v8f acc1 = {};
    #pragma unroll
    for (int i = 0; i < KSTEPS; ++i) {
        const int ka = i * 4 + 2 * hf;
        v2f a0, a1;
        a0.x = xa[ka + 0]; a0.y = xa[ka + 1];
        a1.x = xb[ka + 0]; a1.y = xb[ka + 1];
        acc0 = __builtin_amdgcn_wmma_f32_16x16x4_f32(
                   false, a0, false, bv[i], (short)0, acc0, false, false);
        acc1 = __builtin_amdgcn_wmma_f32_16x16x4_f32(
                   false, a1, false, bv[i], (short)0, acc1, false, false);
    }

    #pragma unroll
    for (int r = 0; r < 8; ++r) {
        const int m = r + hf * 8;
        if (row0 + m < nNodes)
            S[(size_t)(row0 + m) * 16 + l16] = acc0[r];
        if (row0 + 16 + m < nNodes)
            S[(size_t)(row0 + 16 + m) * 16 + l16] = acc1[r];
    }
}

// ---------------------------------------------------------------------------
// Kernel 3: init reduction buffers.
// ---------------------------------------------------------------------------
__global__ void init_kernel(unsigned* __restrict__ mkey,
                            float* __restrict__ ssum,
                            float* __restrict__ num, int n2) {
    int i = blockIdx.x * blockDim.x + threadIdx.x;
    if (i < n2) {
        mkey[i] = NEG_INF_KEY;
        ssum[i] = 0.0f;
        num[i]  = 0.0f;
    }
}

// ---------------------------------------------------------------------------
// Kernel 4: per-edge leaky-relu logit + segment max (atomicMax on ordered key).
// Edges [0,E) come from edge_index (int64); edges [E, E+N) are self loops.
// S columns: 0..1 = a_src[h], 2..3 = a_dst[h], 4..5 = z[h].
// ---------------------------------------------------------------------------
__global__ void edge_max_kernel(const long long* __restrict__ ei, int E, int nNodes,
                                const float* __restrict__ S,
                                unsigned* __restrict__ mkey) {
    int e = blockIdx.x * blockDim.x + threadIdx.x;
    const int total = E + nNodes;
    if (e >= total) return;
    int s, d;
    if (e < E) { s = (int)ei[e]; d = (int)ei[E + e]; }
    else       { s = d = e - E; }
    const float* Ss = S + (size_t)s * 16;
    const float* Sd = S + (size_t)d * 16;
    #pragma unroll
    for (int h = 0; h < NHEAD; ++h) {
        float v = Ss[h] + Sd[2 + h];
        v = (v > 0.0f) ? v : NEG_SLOPE * v;     // leaky_relu
        atomicMax(&mkey[d * NHEAD + h], fkey(v));
    }
}

// ---------------------------------------------------------------------------
// Kernel 5: per-edge softmax numer/denom accumulation.
//   p = exp(e - m[dst]);  ssum[dst,h] += p;  num[dst,h] += p * z[src,h]
// (division by the segment sum is deferred to finalize -> single edge pass)
// ---------------------------------------------------------------------------
__global__ void edge_acc_kernel(const long long* __restrict__ ei, int E, int nNodes,
                                const float* __restrict__ S,
                                const unsigned* __restrict__ mkey,
                                float* __restrict__ ssum,
                                float* __restrict__ num) {
    int e = blockIdx.x * blockDim.x + threadIdx.x;
    const int total = E + nNodes;
    if (e >= total) return;
    int s, d;
    if (e < E) { s = (int)ei[e]; d = (int)ei[E + e]; }
    else       { s = d = e - E; }
    const float* Ss = S + (size_t)s * 16;
    const float* Sd = S + (size_t)d * 16;
    #pragma unroll
    for (int h = 0; h < NHEAD; ++h) {
        float v = Ss[h] + Sd[2 + h];
        v = (v > 0.0f) ? v : NEG_SLOPE * v;
        float m = funkey(mkey[d * NHEAD + h]);   // finite: every dst has a self-loop
        float p = __expf(v - m);
        atomicAdd(&ssum[d * NHEAD + h], p);
        atomicAdd(&num [d * NHEAD + h], p * Ss[4 + h]);
    }
}

// ---------------------------------------------------------------------------
// Kernel 6: finalize  y[n] = mean_h( num/(ssum+1e-16) ) + (bias.W_lin + b_lin)
// ---------------------------------------------------------------------------
__global__ void finalize_kernel(const float* __restrict__ ssum,
                                const float* __restrict__ num,
                                const float* __restrict__ c0,
                                float* __restrict__ y, int nNodes) {
    int n = blockIdx.x * blockDim.x + threadIdx.x;
    if (n >= nNodes) return;
    float acc = 0.0f;
    #pragma unroll
    for (int h = 0; h < NHEAD; ++h)
        acc += num[n * NHEAD + h] / (ssum[n * NHEAD + h] + 1e-16f);
    y[n] = acc * (1.0f / NHEAD) + *c0;
}

// ---------------------------------------------------------------------------
extern "C" void kernel_launch(void* const* d_in, const int* in_sizes, int n_in,
                              void* d_out, int out_size, void* d_ws, size_t ws_size,
                              hipStream_t stream) {
    const float*     x       = (const float*)d_in[0];
    const long long* ei      = (const long long*)d_in[1];
    const float*     W       = (const float*)d_in[2];
    const float*     att_src = (const float*)d_in[3];
    const float*     att_dst = (const float*)d_in[4];
    const float*     bias    = (const float*)d_in[5];
    const float*     W_lin   = (const float*)d_in[6];
    const float*     b_lin   = (const float*)d_in[7];
    float*           y       = (float*)d_out;

    const int nNodes = in_sizes[0] / D_CH;
    const int E      = in_sizes[1] / 2;

    // workspace layout (256B aligned slices)
    char* ws = (char*)d_ws;
    size_t off = 0;
    auto align256 = [](size_t o) { return (o + 255) & ~(size_t)255; };
    float*    U    = (float*)(ws + off); off = align256(off + (size_t)D_CH * 16 * sizeof(float));
    float*    c0   = (float*)(ws + off); off = align256(off + sizeof(float));
    float*    S    = (float*)(ws + off); off = align256(off + (size_t)nNodes * 16 * sizeof(float));
    unsigned* mkey = (unsigned*)(ws + off); off = align256(off + (size_t)nNodes * NHEAD * sizeof(unsigned));
    float*    ssum = (float*)(ws + off); off = align256(off + (size_t)nNodes * NHEAD * sizeof(float));
    float*    num  = (float*)(ws + off); off = align256(off + (size_t)nNodes * NHEAD * sizeof(float));

    prep_U_kernel<<<(D_CH * 16 + 255) / 256, 256, 0, stream>>>(
        W, att_src, att_dst, W_lin, bias, b_lin, U, c0);

    const int pairs = (nNodes + 31) / 32;          // 32-row tile pairs, 8 waves/block
    node_gemm_wmma<<<(pairs + 7) / 8, 256, 0, stream>>>(x, U, S, nNodes);

    const int n2 = nNodes * NHEAD;
    init_kernel<<<(n2 + 255) / 256, 256, 0, stream>>>(mkey, ssum, num, n2);

    const int total = E + nNodes;
    edge_max_kernel<<<(total + 255) / 256, 256, 0, stream>>>(ei, E, nNodes, S, mkey);
    edge_acc_kernel<<<(total + 255) / 256, 256, 0, stream>>>(ei, E, nNodes, S, mkey, ssum, num);

    finalize_kernel<<<(nNodes + 255) / 256, 256, 0, stream>>>(ssum, num, c0, y, nNodes);
}